// MutualInformation_28707561407206
// MI455X (gfx1250) — compile-verified
//
#include <hip/hip_runtime.h>
#include <math.h>

#define BSZ   512
#define DF    256
#define KZ    512   // STEP*DZ = 16*32
#define LOG_2PI 1.8378770664093453f

typedef __attribute__((ext_vector_type(2))) float v2f;
typedef __attribute__((ext_vector_type(8))) float v8f;

// ===========================================================================
// Kernel 1a (staged path): build interleaved B matrices + per-row constants.
//   Bf[j][2k]   = exp(-flv[j,k])            (w)
//   Bf[j][2k+1] = -2 * fm[j,k] * w
//   Cf[j]       = sum_k ( lv + m^2 * w )
// grid = (512, 2): y==0 -> f (K=256), y==1 -> z (K=512) ; block = 256
// ===========================================================================
__global__ void __launch_bounds__(256)
prep_bc_kernel(const float* __restrict__ fm, const float* __restrict__ flv,
               const float* __restrict__ zm, const float* __restrict__ zlv,
               float* __restrict__ Bf, float* __restrict__ Bz,
               float* __restrict__ Cf, float* __restrict__ Cz) {
  __shared__ float red[256];
  const int j = blockIdx.x;
  const int t = threadIdx.x;
  float acc = 0.0f;
  if (blockIdx.y == 0) {
    const float lv = flv[j * DF + t];
    const float m  = fm[j * DF + t];
    const float w  = __expf(-lv);
    ((float2*)Bf)[j * DF + t] = make_float2(w, -2.0f * m * w);
    acc = lv + m * m * w;
  } else {
    #pragma unroll
    for (int k = t; k < KZ; k += 256) {
      const float lv = zlv[j * KZ + k];
      const float m  = zm[j * KZ + k];
      const float w  = __expf(-lv);
      ((float2*)Bz)[j * KZ + k] = make_float2(w, -2.0f * m * w);
      acc += lv + m * m * w;
    }
  }
  red[t] = acc;
  __syncthreads();
  for (int off = 128; off > 0; off >>= 1) {
    if (t < off) red[t] += red[t + off];
    __syncthreads();
  }
  if (t == 0) {
    if (blockIdx.y == 0) Cf[j] = red[0];
    else                 Cz[j] = red[0];
  }
}

// ===========================================================================
// Kernel 1b (fallback path): constants only (no staging workspace needed)
// ===========================================================================
__global__ void __launch_bounds__(256)
prep_c_kernel(const float* __restrict__ fm, const float* __restrict__ flv,
              const float* __restrict__ zm, const float* __restrict__ zlv,
              float* __restrict__ Cf, float* __restrict__ Cz) {
  __shared__ float red[256];
  const int j = blockIdx.x;
  const int t = threadIdx.x;
  float acc = 0.0f;
  if (blockIdx.y == 0) {
    const float lv = flv[j * DF + t];
    const float m  = fm[j * DF + t];
    acc = lv + m * m * __expf(-lv);
  } else {
    #pragma unroll
    for (int k = t; k < KZ; k += 256) {
      const float lv = zlv[j * KZ + k];
      const float m  = zm[j * KZ + k];
      acc += lv + m * m * __expf(-lv);
    }
  }
  red[t] = acc;
  __syncthreads();
  for (int off = 128; off > 0; off >>= 1) {
    if (t < off) red[t] += red[t + off];
    __syncthreads();
  }
  if (t == 0) {
    if (blockIdx.y == 0) Cf[j] = red[0];
    else                 Cz[j] = red[0];
  }
}

// ---------------------------------------------------------------------------
// Shared device helpers for the LSE epilogue
// ---------------------------------------------------------------------------
__device__ __forceinline__ void lse_update(float& m, float& s, float x) {
  const float nm = fmaxf(m, x);
  s = s * __expf(m - nm) + __expf(x - nm);
  m = nm;
}
__device__ __forceinline__ void lse_merge(float& m, float& s, float om, float os) {
  const float nm = fmaxf(m, om);
  s = s * __expf(m - nm) + os * __expf(om - nm);
  m = nm;
}

// ===========================================================================
// Kernel 2 (staged): WMMA GEMM from staged B + on-the-fly A, 4 column tiles
// per wave per pass (A fragment reused across 4 WMMAs), online LSE epilogue.
//   grid = 32 (16-row blocks of i), block = 128 = 4 waves
// ===========================================================================
__global__ void __launch_bounds__(128)
pairwise_lse_staged_kernel(const float* __restrict__ fs,
                           const float* __restrict__ zs,
                           const float* __restrict__ Bf,
                           const float* __restrict__ Bz,
                           const float* __restrict__ Cf,
                           const float* __restrict__ Cz,
                           float* __restrict__ row_lse /* [3][512] */) {
  const int lane = threadIdx.x & 31;
  const int wave = threadIdx.x >> 5;   // 0..3
  const int half = lane >> 4;          // K-phase of this lane's fragments
  const int l16  = lane & 15;
  const int I0   = blockIdx.x * 16;
  const int arow = I0 + l16;

  __shared__ float lds_m[3][4][16];
  __shared__ float lds_s[3][4][16];

  const float* __restrict__ fsrow = fs + arow * DF;
  const float* __restrict__ zsrow = zs + arow * KZ;

  float mq[3][8], sq[3][8];
  #pragma unroll
  for (int st = 0; st < 3; ++st)
    #pragma unroll
    for (int v = 0; v < 8; ++v) { mq[st][v] = -INFINITY; sq[st][v] = 0.0f; }

  #pragma unroll
  for (int grp = 0; grp < 2; ++grp) {
    // 4 consecutive column tiles owned by this wave in this pass
    int jcol[4];
    const v2f* __restrict__ bfrow[4];
    const v2f* __restrict__ bzrow[4];
    #pragma unroll
    for (int c = 0; c < 4; ++c) {
      jcol[c]  = (wave * 8 + grp * 4 + c) * 16 + l16;
      bfrow[c] = (const v2f*)Bf + jcol[c] * DF;   // DF v2f pairs per row
      bzrow[c] = (const v2f*)Bz + jcol[c] * KZ;   // KZ v2f pairs per row
      __builtin_prefetch((const void*)bfrow[c], 0, 0);
      __builtin_prefetch((const void*)bzrow[c], 0, 0);
    }

    // ---- f domain: 128 K-steps, 4 WMMAs each ----
    v8f accf[4] = {};
    #pragma unroll 2
    for (int t = 0; t < DF / 2; ++t) {
      const float sv = fsrow[2 * t + half];
      v2f a; a[0] = sv * sv; a[1] = sv;
      #pragma unroll
      for (int c = 0; c < 4; ++c) {
        const v2f b = bfrow[c][2 * t + half];
        accf[c] = __builtin_amdgcn_wmma_f32_16x16x4_f32(
            false, a, false, b, (short)0, accf[c], false, false);
      }
    }

    // ---- z domain: 256 K-steps, 4 WMMAs each ----
    v8f accz[4] = {};
    #pragma unroll 2
    for (int t = 0; t < KZ / 2; ++t) {
      const float sv = zsrow[2 * t + half];
      v2f a; a[0] = sv * sv; a[1] = sv;
      #pragma unroll
      for (int c = 0; c < 4; ++c) {
        const v2f b = bzrow[c][2 * t + half];
        accz[c] = __builtin_amdgcn_wmma_f32_16x16x4_f32(
            false, a, false, b, (short)0, accz[c], false, false);
      }
    }

    // ---- fold constants, online LSE over the 4 tiles ----
    #pragma unroll
    for (int c = 0; c < 4; ++c) {
      const float cfj = Cf[jcol[c]];
      const float czj = Cz[jcol[c]];
      #pragma unroll
      for (int v = 0; v < 8; ++v) {
        const float gf = -0.5f * (accf[c][v] + cfj);
        const float gz = -0.5f * (accz[c][v] + czj);
        lse_update(mq[0][v], sq[0][v], gf);
        lse_update(mq[1][v], sq[1][v], gz);
        lse_update(mq[2][v], sq[2][v], gf + gz);
      }
    }
  }

  // ---- reduce across the 16 lanes of each half-wave (columns) ----
  #pragma unroll
  for (int st = 0; st < 3; ++st) {
    #pragma unroll
    for (int v = 0; v < 8; ++v) {
      float m = mq[st][v], s = sq[st][v];
      #pragma unroll
      for (int off = 8; off >= 1; off >>= 1) {
        lse_merge(m, s, __shfl_xor(m, off, 32), __shfl_xor(s, off, 32));
      }
      if (l16 == 0) {
        const int row = v + 8 * half;
        lds_m[st][wave][row] = m;
        lds_s[st][wave][row] = s;
      }
    }
  }
  __syncthreads();

  if (threadIdx.x < 16) {
    const int r = threadIdx.x;
    const float cst[3] = { -0.5f * DF * LOG_2PI,
                           -0.5f * KZ * LOG_2PI,
                           -0.5f * (DF + KZ) * LOG_2PI };
    #pragma unroll
    for (int st = 0; st < 3; ++st) {
      float m = lds_m[st][0][r], s = lds_s[st][0][r];
      #pragma unroll
      for (int w2 = 1; w2 < 4; ++w2)
        lse_merge(m, s, lds_m[st][w2][r], lds_s[st][w2][r]);
      row_lse[st * BSZ + I0 + r] = m + __logf(s) + cst[st];
    }
  }
}

// ===========================================================================
// Kernel 2' (fallback, no staging): fused exp recompute
// ===========================================================================
__global__ void __launch_bounds__(128)
pairwise_lse_fused_kernel(const float* __restrict__ fm, const float* __restrict__ flv,
                          const float* __restrict__ fs,
                          const float* __restrict__ zm, const float* __restrict__ zlv,
                          const float* __restrict__ zs,
                          const float* __restrict__ Cf, const float* __restrict__ Cz,
                          float* __restrict__ row_lse) {
  const int lane = threadIdx.x & 31;
  const int wave = threadIdx.x >> 5;
  const int half = lane >> 4;
  const int l16  = lane & 15;
  const int I0   = blockIdx.x * 16;
  const int arow = I0 + l16;

  __shared__ float lds_m[3][4][16];
  __shared__ float lds_s[3][4][16];

  float mq[3][8], sq[3][8];
  #pragma unroll
  for (int st = 0; st < 3; ++st)
    #pragma unroll
    for (int v = 0; v < 8; ++v) { mq[st][v] = -INFINITY; sq[st][v] = 0.0f; }

  for (int jt = wave; jt < 32; jt += 4) {
    const int jcol = jt * 16 + l16;

    v8f accf = {};
    {
      const float* __restrict__ srow = fs  + arow * DF;
      const float* __restrict__ lrow = flv + jcol * DF;
      const float* __restrict__ mrow = fm  + jcol * DF;
      #pragma unroll 4
      for (int t = 0; t < DF / 2; ++t) {
        const int k = 2 * t + half;
        const float sv = srow[k];
        const float lv = lrow[k];
        const float mv = mrow[k];
        const float w  = __expf(-lv);
        v2f a; a[0] = sv * sv; a[1] = sv;
        v2f b; b[0] = w;       b[1] = -2.0f * mv * w;
        accf = __builtin_amdgcn_wmma_f32_16x16x4_f32(
            false, a, false, b, (short)0, accf, false, false);
      }
    }

    v8f accz = {};
    {
      const float* __restrict__ srow = zs  + arow * KZ;
      const float* __restrict__ lrow = zlv + jcol * KZ;
      const float* __restrict__ mrow = zm  + jcol * KZ;
      #pragma unroll 4
      for (int t = 0; t < KZ / 2; ++t) {
        const int k = 2 * t + half;
        const float sv = srow[k];
        const float lv = lrow[k];
        const float mv = mrow[k];
        const float w  = __expf(-lv);
        v2f a; a[0] = sv * sv; a[1] = sv;
        v2f b; b[0] = w;       b[1] = -2.0f * mv * w;
        accz = __builtin_amdgcn_wmma_f32_16x16x4_f32(
            false, a, false, b, (short)0, accz, false, false);
      }
    }

    const float cfj = Cf[jcol];
    const float czj = Cz[jcol];
    #pragma unroll
    for (int v = 0; v < 8; ++v) {
      const float gf = -0.5f * (accf[v] + cfj);
      const float gz = -0.5f * (accz[v] + czj);
      lse_update(mq[0][v], sq[0][v], gf);
      lse_update(mq[1][v], sq[1][v], gz);
      lse_update(mq[2][v], sq[2][v], gf + gz);
    }
  }

  #pragma unroll
  for (int st = 0; st < 3; ++st) {
    #pragma unroll
    for (int v = 0; v < 8; ++v) {
      float m = mq[st][v], s = sq[st][v];
      #pragma unroll
      for (int off = 8; off >= 1; off >>= 1)
        lse_merge(m, s, __shfl_xor(m, off, 32), __shfl_xor(s, off, 32));
      if (l16 == 0) {
        const int row = v + 8 * half;
        lds_m[st][wave][row] = m;
        lds_s[st][wave][row] = s;
      }
    }
  }
  __syncthreads();

  if (threadIdx.x < 16) {
    const int r = threadIdx.x;
    const float cst[3] = { -0.5f * DF * LOG_2PI,
                           -0.5f * KZ * LOG_2PI,
                           -0.5f * (DF + KZ) * LOG_2PI };
    #pragma unroll
    for (int st = 0; st < 3; ++st) {
      float m = lds_m[st][0][r], s = lds_s[st][0][r];
      #pragma unroll
      for (int w2 = 1; w2 < 4; ++w2)
        lse_merge(m, s, lds_m[st][w2][r], lds_s[st][w2][r]);
      row_lse[st * BSZ + I0 + r] = m + __logf(s) + cst[st];
    }
  }
}

// ===========================================================================
// Kernel 3: means over rows + entropy / MI scalars
// ===========================================================================
__global__ void __launch_bounds__(512)
finalize_kernel(const float* __restrict__ row_lse,
                const int* __restrict__ num_train,
                float* __restrict__ out) {
  __shared__ float red[3][512];
  const int t = threadIdx.x;
  #pragma unroll
  for (int st = 0; st < 3; ++st) red[st][t] = row_lse[st * BSZ + t];
  __syncthreads();
  for (int off = 256; off > 0; off >>= 1) {
    if (t < off) {
      #pragma unroll
      for (int st = 0; st < 3; ++st) red[st][t] += red[st][t + off];
    }
    __syncthreads();
  }
  if (t == 0) {
    const float logNM = __logf((float)num_train[0] * (float)BSZ);
    const float Hf  = -red[0][0] / (float)BSZ + logNM;
    const float Hz  = -red[1][0] / (float)BSZ + logNM;
    const float Hfz = -red[2][0] / (float)BSZ + logNM;
    out[0] = Hf + Hz - Hfz;   // mutual information
    out[1] = Hf;
    out[2] = Hz;
    out[3] = Hfz;
  }
}

// ===========================================================================
extern "C" void kernel_launch(void* const* d_in, const int* in_sizes, int n_in,
                              void* d_out, int out_size, void* d_ws, size_t ws_size,
                              hipStream_t stream) {
  const float* fm  = (const float*)d_in[0];
  const float* flv = (const float*)d_in[1];
  const float* fs  = (const float*)d_in[2];
  const float* zm  = (const float*)d_in[3];
  const float* zlv = (const float*)d_in[4];
  const float* zs  = (const float*)d_in[5];
  const int*   ntr = (const int*)d_in[6];

  float* ws      = (float*)d_ws;
  float* Cf      = ws;                 // 512
  float* Cz      = ws + BSZ;           // 512
  float* row_lse = ws + 2 * BSZ;       // 3*512
  float* Bf      = ws + 2 * BSZ + 3 * BSZ;          // 512*512   (offset 2560)
  float* Bz      = Bf + BSZ * (2 * DF);             // 512*1024

  const size_t ws_needed =
      (size_t)(2 * BSZ + 3 * BSZ + BSZ * (2 * DF) + BSZ * (2 * KZ)) * sizeof(float);

  dim3 gridC(BSZ, 2);
  if (ws_size >= ws_needed) {
    prep_bc_kernel<<<gridC, 256, 0, stream>>>(fm, flv, zm, zlv, Bf, Bz, Cf, Cz);
    pairwise_lse_staged_kernel<<<32, 128, 0, stream>>>(fs, zs, Bf, Bz, Cf, Cz,
                                                       row_lse);
  } else {
    prep_c_kernel<<<gridC, 256, 0, stream>>>(fm, flv, zm, zlv, Cf, Cz);
    pairwise_lse_fused_kernel<<<32, 128, 0, stream>>>(fm, flv, fs, zm, zlv, zs,
                                                      Cf, Cz, row_lse);
  }
  finalize_kernel<<<1, 512, 0, stream>>>(row_lse, ntr, (float*)d_out);
}